// AudioFrontend_51814485459236
// MI455X (gfx1250) — compile-verified
//
#include <hip/hip_runtime.h>
#include <math.h>

#define N_PDM_TOT 10080000
#define NDEC      160000
#define NTAPS     15
#define NFFT      512
#define HOP       160
#define WINLEN    400
#define WINPAD    56          // (512-400)/2
#define NFREQ     257
#define NFREQ_P   272         // 17*16
#define NMELS     40
#define NMELS_P   48          // 3*16
#define NFRAMES   997
#define NFRAMES_P 1008        // 63*16
#define CIC_LEN   311         // 5*62+1

typedef __attribute__((ext_vector_type(2))) float v2f;
typedef __attribute__((ext_vector_type(8))) float v8f;

// ---------------------------------------------------------------------------
// Kernel 1: exact-integer CIC. 5x cumsum -> ::63 -> 5x diff  ==  311-tap
// integer FIR (5-fold convolution of ones(63)) evaluated at stride 63.
// ---------------------------------------------------------------------------
__global__ void __launch_bounds__(256) k_cic(const int* __restrict__ bits,
                                             float* __restrict__ y) {
  __shared__ signed char sx[63 * 255 + 311];   // 16376 staged +/-1 samples
  __shared__ int h[CIC_LEN];
  const int tid = threadIdx.x;

  // Build h[k] = sum_j (-1)^j C(5,j) * C(k-63j+4, 4)   (exact, int64 interm.)
  for (int k = tid; k < CIC_LEN; k += 256) {
    long long acc = 0;
    long long c5 = 1;                          // C(5,j)
    for (int j = 0; j <= 5; ++j) {
      long long n = (long long)k - 63LL * j;
      if (n >= 0) {
        long long b5 = (n + 4) * (n + 3) * (n + 2) * (n + 1) / 24;
        acc += ((j & 1) ? -c5 : c5) * b5;
      }
      c5 = c5 * (5 - j) / (j + 1);
    }
    h[k] = (int)acc;
  }

  const int m0 = blockIdx.x * 256;
  const long long start = (long long)m0 * 63 - 310;
  const int span = 63 * 255 + 311;
  for (int i = tid; i < span; i += 256) {
    long long g = start + i;
    int v = 0;
    if (g >= 0 && g < N_PDM_TOT) v = 2 * bits[g] - 1;
    sx[i] = (signed char)v;
  }
  __syncthreads();

  int acc = 0;                                 // |acc| <= 63^5 < 2^31: exact
  const int base = 63 * tid;
  #pragma unroll 4
  for (int j = 0; j < CIC_LEN; ++j)
    acc += h[310 - j] * (int)sx[base + j];

  const int m = m0 + tid;
  if (m < NDEC) y[m] = (float)acc;
}

// ---------------------------------------------------------------------------
// Kernel 2: 15-tap compensation FIR: s[i] = sum_d taps[d] * y[i+d-7]
// (== jnp.convolve(y, taps[::-1], 'same'))
// ---------------------------------------------------------------------------
__global__ void __launch_bounds__(256) k_fir(const float* __restrict__ y,
                                             const float* __restrict__ taps,
                                             float* __restrict__ s) {
  const int i = blockIdx.x * 256 + threadIdx.x;
  if (i >= NDEC) return;
  float acc = 0.f;
  #pragma unroll
  for (int d = 0; d < NTAPS; ++d) {
    const int j = i + d - 7;
    const float v = (j >= 0 && j < NDEC) ? y[j] : 0.f;
    acc = fmaf(taps[d], v, acc);
  }
  s[i] = acc;
}

// ---------------------------------------------------------------------------
// Kernel 3a: window-folded rDFT matrices, stored K-contiguous (transposed):
//   BreT[f][n] =  win[n]*cos(2*pi*n*f/512)
//   BimT[f][n] = -win[n]*sin(2*pi*n*f/512)     (rows f>=257 are zero)
// ---------------------------------------------------------------------------
__global__ void __launch_bounds__(256) k_dftmat(const float* __restrict__ window,
                                                float* __restrict__ BreT,
                                                float* __restrict__ BimT) {
  const int id = blockIdx.x * 256 + threadIdx.x;
  if (id >= NFREQ_P * NFFT) return;
  const int f = id / NFFT;
  const int n = id % NFFT;
  float wr = 0.f, wi = 0.f;
  if (f < NFREQ) {
    const float w = (n >= WINPAD && n < WINPAD + WINLEN) ? window[n - WINPAD] : 0.f;
    const int ph = (n * f) & (NFFT - 1);       // arg reduction, exact in int
    const float ang = 1.2271846644580963e-2f * (float)ph;  // 2*pi/512
    wr =  w * cosf(ang);
    wi = -w * sinf(ang);
  }
  BreT[id] = wr;
  BimT[id] = wi;
}

// ---------------------------------------------------------------------------
// Kernel 3b: mel filterbank transposed + zero padded: fbT[48][272]
//   fbT[m][f] = mel_fb[f][m]  (m<40, f<257), else 0
// ---------------------------------------------------------------------------
__global__ void __launch_bounds__(256) k_fbt(const float* __restrict__ fb,
                                             float* __restrict__ fbT) {
  const int id = blockIdx.x * 256 + threadIdx.x;
  if (id >= NMELS_P * NFREQ_P) return;
  const int m = id / NFREQ_P;
  const int f = id % NFREQ_P;
  fbT[id] = (m < NMELS && f < NFREQ) ? fb[f * NMELS + m] : 0.f;
}

// ---------------------------------------------------------------------------
// Kernel 4: DFT GEMM + power. One wave32 per 16x16 (t,f) tile; EXEC all-ones.
//   Re(997x257) = frames(997x512) * B ; magT[t][f] = Re^2 + Im^2
// Pad rows t>=997 clamp to row 996 (harmless: only feeds discarded outputs).
// All operand fetches are aligned global_load_b64 (K-pairs contiguous).
// ---------------------------------------------------------------------------
__global__ void __launch_bounds__(32) k_dft_gemm(const float* __restrict__ s,
                                                 const float* __restrict__ BreT,
                                                 const float* __restrict__ BimT,
                                                 float* __restrict__ magT) {
  const int lane = threadIdx.x;
  const int half = lane >> 4, l16 = lane & 15;
  const int t0 = blockIdx.x * 16;              // frame tile
  const int f0 = blockIdx.y * 16;              // freq tile
  const int tA = min(t0 + l16, NFRAMES - 1);   // clamped A row
  const int f  = f0 + l16;

  const float* arow = s + tA * HOP;            // A[M=l16, K]
  const float* rrow = BreT + f * NFFT;         // B[K, N=l16] (K-contiguous)
  const float* irow = BimT + f * NFFT;

  v8f accR = {};
  v8f accI = {};
  #pragma unroll 8
  for (int k0 = 0; k0 < NFFT; k0 += 4) {
    const int kr = k0 + 2 * half;              // this lane's K pair: kr, kr+1
    const v2f a  = *(const v2f*)(arow + kr);
    const v2f br = *(const v2f*)(rrow + kr);
    const v2f bi = *(const v2f*)(irow + kr);
    accR = __builtin_amdgcn_wmma_f32_16x16x4_f32(false, a, false, br,
                                                 (short)0, accR, false, false);
    accI = __builtin_amdgcn_wmma_f32_16x16x4_f32(false, a, false, bi,
                                                 (short)0, accI, false, false);
  }
  #pragma unroll
  for (int r = 0; r < 8; ++r) {                // D[M=8*half+r, N=l16]
    const int tt = t0 + 8 * half + r;
    const float p = accR[r] * accR[r] + accI[r] * accI[r];
    magT[tt * NFREQ_P + f] = p;                // [t][f]: K-contiguous for mel
  }
}

// ---------------------------------------------------------------------------
// Kernel 5: mel GEMM + log. out[m,t] = log(sum_f magT[t][f]*fbT[m][f] + 1e-6)
// A = fbT (48x272 zero-padded), B = magT^T via K-contiguous b64 fetches.
// ---------------------------------------------------------------------------
__global__ void __launch_bounds__(32) k_mel_gemm(const float* __restrict__ magT,
                                                 const float* __restrict__ fbT,
                                                 float* __restrict__ out) {
  const int lane = threadIdx.x;
  const int half = lane >> 4, l16 = lane & 15;
  const int m0 = blockIdx.x * 16;
  const int t0 = blockIdx.y * 16;

  const float* arow = fbT + (m0 + l16) * NFREQ_P;   // A[M=l16, K]
  const float* brow = magT + (t0 + l16) * NFREQ_P;  // B[K, N=l16]

  v8f acc = {};
  #pragma unroll 5
  for (int k0 = 0; k0 < 260; k0 += 4) {        // K = 257 padded to 260
    const int kr = k0 + 2 * half;              // cols 257..259 of fbT are zero
    const v2f a = *(const v2f*)(arow + kr);
    const v2f b = *(const v2f*)(brow + kr);
    acc = __builtin_amdgcn_wmma_f32_16x16x4_f32(false, a, false, b,
                                                (short)0, acc, false, false);
  }
  #pragma unroll
  for (int r = 0; r < 8; ++r) {
    const int m = m0 + 8 * half + r;
    const int t = t0 + l16;
    if (m < NMELS && t < NFRAMES)
      out[m * NFRAMES + t] = logf(acc[r] + 1e-6f);
  }
}

// ---------------------------------------------------------------------------
extern "C" void kernel_launch(void* const* d_in, const int* in_sizes, int n_in,
                              void* d_out, int out_size, void* d_ws, size_t ws_size,
                              hipStream_t stream) {
  const int*   bits   = (const int*)d_in[0];     // pdm_bits (int32 0/1)
  const float* taps   = (const float*)d_in[1];   // 15 floats
  const float* window = (const float*)d_in[2];   // 400 floats
  const float* mel_fb = (const float*)d_in[3];   // 257*40 floats
  float* out = (float*)d_out;                    // 40*997 floats

  float* ws   = (float*)d_ws;
  float* y    = ws;                              // 160000
  float* s    = y + NDEC;                        // 160000
  float* BreT = s + NDEC;                        // 272*512
  float* BimT = BreT + NFREQ_P * NFFT;           // 272*512
  float* fbT  = BimT + NFREQ_P * NFFT;           // 48*272
  float* magT = fbT + NMELS_P * NFREQ_P;         // 1008*272

  k_cic<<<NDEC / 256, 256, 0, stream>>>(bits, y);
  k_fir<<<(NDEC + 255) / 256, 256, 0, stream>>>(y, taps, s);
  k_dftmat<<<(NFREQ_P * NFFT + 255) / 256, 256, 0, stream>>>(window, BreT, BimT);
  k_fbt<<<(NMELS_P * NFREQ_P + 255) / 256, 256, 0, stream>>>(mel_fb, fbT);
  k_dft_gemm<<<dim3(63, 17), 32, 0, stream>>>(s, BreT, BimT, magT);
  k_mel_gemm<<<dim3(3, 63), 32, 0, stream>>>(magT, fbT, out);
}